// GeneralScatter_24223615549678
// MI455X (gfx1250) — compile-verified
//
#include <hip/hip_runtime.h>
#include <stdint.h>

// Problem constants (fixed by the reference).
#define NY   496
#define NX   432
#define NZ   1
#define CCH  64
#define NVOX 320000
#define BATCH 4
#define SPC  (NY * NX * NZ)          // 214272 spatial cells per batch image
#define MAP_ELEMS (BATCH * SPC)      // 857088 cells total
#define OUT_ELEMS (BATCH * CCH * SPC) // 54,853,632 output floats (~219 MB)

typedef float v4f __attribute__((ext_vector_type(4)));
typedef int   v4i __attribute__((ext_vector_type(4)));

// ---------------------------------------------------------------------------
// Phase 0: map[cell] = -1 for all cells. Regular (RT) stores: the map is hot,
// we want it resident in L2 for phases 1 and 2 (3.4 MB << 192 MB L2).
// ---------------------------------------------------------------------------
__global__ void ps_init_map(int* __restrict__ map) {
    int t = blockIdx.x * blockDim.x + threadIdx.x;   // one int4 per thread
    if (t < MAP_ELEMS / 4) {
        v4i m = { -1, -1, -1, -1 };
        ((v4i*)map)[t] = m;
    }
}

// ---------------------------------------------------------------------------
// Phase 1: map[spatial(b,z,y,x)] = max over voxels mapping there.
// global_atomic_max_i32 makes the duplicate-winner deterministic.
// coors rows are int4 -> coalesced b128 loads, read once -> non-temporal.
// ---------------------------------------------------------------------------
__global__ void ps_build_map(const int* __restrict__ coors,
                             int* __restrict__ map) {
    int n = blockIdx.x * blockDim.x + threadIdx.x;
    if (n >= NVOX) return;
    v4i c = __builtin_nontemporal_load((const v4i*)coors + n);
    int b = c[0], z = c[1], y = c[2], x = c[3];
    int g = b * SPC + y * (NX * NZ) + x * NZ + z;
    atomicMax(&map[g], n);
}

// ---------------------------------------------------------------------------
// Phase 2: inverted gather. Thread t owns 4 consecutive x-cells of one (b,y)
// row. It reads 4 map entries (b128, L2-hot), prefetches the 4 feature rows
// (global_prefetch_b8), then for each channel-quad performs a 4x4 register
// transpose: 4x b128 NT loads walking each 256 B feature row sequentially,
// and 4x b128 NT stores that are contiguous across the wave (128 consecutive
// x per wave instruction). Empty cells emit zeros, so the output is written
// exactly once and never read -> TH=NT keeps it out of L2.
// Output layout: out[b][c][y][x][z] = out[b*C*SPC + c*SPC + y*NX*NZ + x*NZ + z]
// ---------------------------------------------------------------------------
__global__ void ps_gather_out(const float* __restrict__ vf,
                              const int* __restrict__ map,
                              float* __restrict__ out) {
    int t = blockIdx.x * blockDim.x + threadIdx.x;   // [0, MAP_ELEMS/4)
    if (t >= MAP_ELEMS / 4) return;

    int sbase = t * 4;                 // global spatial cell id (16B aligned)
    int b  = sbase / SPC;              // const-div -> mul/shift
    int sl = sbase - b * SPC;          // y*NX + x within the batch image

    v4i m = *(const v4i*)(map + sbase);

    const v4f* r0 = (const v4f*)(vf + (size_t)(m[0] < 0 ? 0 : m[0]) * CCH);
    const v4f* r1 = (const v4f*)(vf + (size_t)(m[1] < 0 ? 0 : m[1]) * CCH);
    const v4f* r2 = (const v4f*)(vf + (size_t)(m[2] < 0 ? 0 : m[2]) * CCH);
    const v4f* r3 = (const v4f*)(vf + (size_t)(m[3] < 0 ? 0 : m[3]) * CCH);
    if (m[0] >= 0) __builtin_prefetch(r0, 0, 0);   // global_prefetch_b8
    if (m[1] >= 0) __builtin_prefetch(r1, 0, 0);
    if (m[2] >= 0) __builtin_prefetch(r2, 0, 0);
    if (m[3] >= 0) __builtin_prefetch(r3, 0, 0);

    float* obase = out + (size_t)b * (CCH * SPC) + sl;
    const v4f vz = { 0.f, 0.f, 0.f, 0.f };

#pragma unroll 4
    for (int cg = 0; cg < CCH / 4; ++cg) {         // channel quads
        v4f a0 = (m[0] >= 0) ? __builtin_nontemporal_load(r0 + cg) : vz;
        v4f a1 = (m[1] >= 0) ? __builtin_nontemporal_load(r1 + cg) : vz;
        v4f a2 = (m[2] >= 0) ? __builtin_nontemporal_load(r2 + cg) : vz;
        v4f a3 = (m[3] >= 0) ? __builtin_nontemporal_load(r3 + cg) : vz;
#pragma unroll
        for (int j = 0; j < 4; ++j) {              // 4x4 register transpose
            v4f w = { a0[j], a1[j], a2[j], a3[j] };
            __builtin_nontemporal_store(
                w, (v4f*)(obase + (size_t)(cg * 4 + j) * SPC));
        }
    }
}

// ---------------------------------------------------------------------------
// Fallback path (only if ws_size < 3.4 MB, which should not happen):
// zero-fill + direct scatter. Note: duplicate voxels race here (reference's
// duplicate winner is unspecified too); primary path is fully deterministic.
// ---------------------------------------------------------------------------
__global__ void ps_zero_out(float* __restrict__ out) {
    int t = blockIdx.x * blockDim.x + threadIdx.x;
    if (t < OUT_ELEMS / 4) {
        v4f vz = { 0.f, 0.f, 0.f, 0.f };
        __builtin_nontemporal_store(vz, (v4f*)out + t);
    }
}

__global__ void ps_scatter_direct(const float* __restrict__ vf,
                                  const int* __restrict__ coors,
                                  float* __restrict__ out) {
    int tid = blockIdx.x * blockDim.x + threadIdx.x;  // 16 threads per voxel
    int n = tid >> 4;
    int q = tid & 15;                                 // channel quad
    if (n >= NVOX) return;
    v4i c = __builtin_nontemporal_load((const v4i*)coors + n);
    int b = c[0], z = c[1], y = c[2], x = c[3];
    int sl = y * (NX * NZ) + x * NZ + z;
    v4f a = ((const v4f*)(vf + (size_t)n * CCH))[q];  // coalesced b128 read
    float* o = out + (size_t)b * (CCH * SPC) + sl;
#pragma unroll
    for (int j = 0; j < 4; ++j)
        __builtin_nontemporal_store(a[j], o + (size_t)(q * 4 + j) * SPC);
}

// ---------------------------------------------------------------------------
extern "C" void kernel_launch(void* const* d_in, const int* in_sizes, int n_in,
                              void* d_out, int out_size, void* d_ws,
                              size_t ws_size, hipStream_t stream) {
    const float* vf    = (const float*)d_in[0];   // (N, 64) fp32
    const int*   coors = (const int*)d_in[1];     // (N, 4) int32
    // d_in[2] = batch_size scalar; fixed to 4 by the reference constants.
    float* out = (float*)d_out;

    const size_t map_bytes = (size_t)MAP_ELEMS * sizeof(int);

    if (ws_size >= map_bytes) {
        int* map = (int*)d_ws;
        ps_init_map<<<(MAP_ELEMS / 4 + 255) / 256, 256, 0, stream>>>(map);
        ps_build_map<<<(NVOX + 255) / 256, 256, 0, stream>>>(coors, map);
        ps_gather_out<<<(MAP_ELEMS / 4 + 255) / 256, 256, 0, stream>>>(
            vf, map, out);
    } else {
        ps_zero_out<<<(OUT_ELEMS / 4 + 255) / 256, 256, 0, stream>>>(out);
        ps_scatter_direct<<<(NVOX * 16 + 255) / 256, 256, 0, stream>>>(
            vf, coors, out);
    }
}